// WaveletBlock1D_4973572129293
// MI455X (gfx1250) — compile-verified
//
#include <hip/hip_runtime.h>
#include <hip/hip_bf16.h>
#include <math.h>

// ---------------------------------------------------------------------------
// Problem constants (from reference): B=32, N=8192, C=64, db4 (L=8), 4 levels
// ---------------------------------------------------------------------------
#define BATCH 32
#define NLEN  8192
#define CH    64
#define MROWS (BATCH * CH)      // 2048 independent signals
#define M1 4099                 // (8192+7)/2
#define M2 2053
#define M3 1030
#define M4 518

typedef float v2f __attribute__((ext_vector_type(2)));
typedef float v8f __attribute__((ext_vector_type(8)));

// db4 filters (match reference _DEC_LO/_DEC_HI; REC = reversed DEC)
__device__ __constant__ float DEC_LO[8] = {
    -0.010597401784997278f,  0.032883011666982945f,  0.030841381835986965f,
    -0.18703481171888114f,  -0.02798376941698385f,   0.6308807679295904f,
     0.7148465705525415f,    0.23037781330885523f };
__device__ __constant__ float DEC_HI[8] = {
    -0.23037781330885523f,   0.7148465705525415f,   -0.6308807679295904f,
    -0.02798376941698385f,   0.18703481171888114f,   0.030841381835986965f,
    -0.032883011666982945f, -0.010597401784997278f };
__device__ __constant__ float REC_LO[8] = {
     0.23037781330885523f,   0.7148465705525415f,    0.6308807679295904f,
    -0.02798376941698385f,  -0.18703481171888114f,   0.030841381835986965f,
     0.032883011666982945f, -0.010597401784997278f };
__device__ __constant__ float REC_HI[8] = {
    -0.010597401784997278f, -0.032883011666982945f,  0.030841381835986965f,
     0.18703481171888114f,  -0.02798376941698385f,  -0.6308807679295904f,
     0.7148465705525415f,   -0.23037781330885523f };

__device__ __forceinline__ int symi(int i, int n) {
    // half-sample symmetric reflection, single bounce (|offset| <= 7 << n)
    if (i < 0)  i = -1 - i;
    if (i >= n) i = 2 * n - 1 - i;
    return i;
}

// ---------------------------------------------------------------------------
// (B,N,C) -> (B*C, N) transpose, LDS-tiled so both sides are coalesced
// ---------------------------------------------------------------------------
__global__ void transpose_in_kernel(const float* __restrict__ x, float* __restrict__ xc) {
    __shared__ float tile[32][33];
    const int b  = blockIdx.z;
    const int n0 = blockIdx.x * 32;
    const int c0 = blockIdx.y * 32;
#pragma unroll
    for (int i = 0; i < 4; ++i) {
        int nn = n0 + threadIdx.y + i * 8;
        tile[threadIdx.y + i * 8][threadIdx.x] =
            x[(b * NLEN + nn) * CH + c0 + threadIdx.x];
    }
    __syncthreads();
#pragma unroll
    for (int i = 0; i < 4; ++i) {
        int cc = c0 + threadIdx.y + i * 8;
        xc[(b * CH + cc) * NLEN + n0 + threadIdx.x] =
            tile[threadIdx.x][threadIdx.y + i * 8];
    }
}

// ---------------------------------------------------------------------------
// One DWT level: in (MROWS x n) -> cA, cD (MROWS x m); y[k] = sum h[s] x_sym[2k+1-s]
// ---------------------------------------------------------------------------
__global__ void dwt_kernel(const float* __restrict__ in, float* __restrict__ cA,
                           float* __restrict__ cD, int n, int m) {
    const int idx = blockIdx.x * blockDim.x + threadIdx.x;   // < MROWS*m <= 8.4M
    if (idx >= MROWS * m) return;
    const int k   = idx % m;
    const int row = idx / m;
    const float* xr = in + row * n;
    float lo = 0.f, hi = 0.f;
    const int base = 2 * k + 1;
#pragma unroll
    for (int s = 0; s < 8; ++s) {
        float v = xr[symi(base - s, n)];
        lo = fmaf(DEC_LO[s], v, lo);
        hi = fmaf(DEC_HI[s], v, hi);
    }
    cA[idx] = lo;
    cD[idx] = hi;
}

// ---------------------------------------------------------------------------
// One IDWT level: a[i] = sum_{s: q=i+6-s even, 0<=q/2<m} REC_LO[s]*ca[q/2]+REC_HI[s]*cd[q/2]
// ---------------------------------------------------------------------------
__global__ void idwt_kernel(const float* __restrict__ ca, const float* __restrict__ cd,
                            int m, float* __restrict__ out, int n) {
    const int idx = blockIdx.x * blockDim.x + threadIdx.x;   // < MROWS*n <= 16.8M
    if (idx >= MROWS * n) return;
    const int i   = idx % n;
    const int row = idx / n;
    const float* car = ca + row * m;
    const float* cdr = cd + row * m;
    float acc = 0.f;
#pragma unroll
    for (int s = 0; s < 8; ++s) {
        int q = i + 6 - s;
        if ((q & 1) == 0) {
            int k = q >> 1;
            if (k >= 0 && k < m)
                acc = fmaf(REC_LO[s], car[k], fmaf(REC_HI[s], cdr[k], acc));
        }
    }
    out[idx] = acc;
}

// ---------------------------------------------------------------------------
// Channel mixing: out[b,o,pos] = sum_i src[b,i,pos] * w[i,o,pos]  (b=32, i=o=64)
// One block (8 waves) per m-position. The per-position A (32x64) and B (64x64)
// slices are staged once into LDS (24 KB), then 8 waves run 16 WMMA K-steps of
// V_WMMA_F32_16X16X4_F32 fed from LDS. m = M4 = 518 is compile-time so all
// staging offsets fold into instruction offsets. EXEC all-ones throughout.
// ---------------------------------------------------------------------------
__global__ void mix_wmma_kernel(const float* __restrict__ src, const float* __restrict__ w,
                                float* __restrict__ dst) {
    __shared__ float sA[BATCH * CH];   // [b*64 + i]
    __shared__ float sB[CH * CH];      // [i*64 + o]
    const int pos = blockIdx.x;
    const int tid = threadIdx.x;

    // Stage A: 2048 values, 8 per thread (gathers along the m-strided layout)
#pragma unroll
    for (int j = 0; j < (BATCH * CH) / 256; ++j) {
        const int e = tid + j * 256;
        sA[e] = src[e * M4 + pos];
    }
    // Stage B: 4096 values, 16 per thread
#pragma unroll
    for (int j = 0; j < (CH * CH) / 256; ++j) {
        const int e = tid + j * 256;
        sB[e] = w[e * M4 + pos];
    }
    __syncthreads();

    const int wave = tid >> 5;
    const int lane = tid & 31;
    const int rt = wave >> 2;          // 0..1 (b-tile)
    const int ct = wave & 3;           // 0..3 (o-tile)
    const int l15   = lane & 15;
    const int khalf = (lane >> 4) * 2;
    const int brow = rt * 16 + l15;    // A row (b index)
    const int ocol = ct * 16 + l15;    // B col (o index)

    v8f acc = {};
#pragma unroll
    for (int k0 = 0; k0 < CH; k0 += 4) {
        const int ka = k0 + khalf;                       // even -> 8B aligned
        const v2f a = *(const v2f*)&sA[brow * CH + ka];  // K=ka, ka+1
        v2f bf;
        bf.x = sB[ka * CH + ocol];
        bf.y = sB[(ka + 1) * CH + ocol];
        acc = __builtin_amdgcn_wmma_f32_16x16x4_f32(false, a, false, bf,
                                                    (short)0, acc, false, false);
    }
    const int bbase = rt * 16 + (lane >> 4) * 8;   // C/D layout: VGPR r -> M=r (+8 hi lanes)
#pragma unroll
    for (int r = 0; r < 8; ++r)
        dst[((bbase + r) * CH + ocol) * M4 + pos] = acc[r];
}

// ---------------------------------------------------------------------------
// Fused shortcut + add + mish:
//   out[row, c] = mish( (x @ K)[row, c] + bias[c] + yrec[(b*C + c)*N + n] )
// row = b*N + n over 262144 rows. Dense kernel staged in LDS (same for every
// block). One wave per 16x64 tile, 4 f32 accumulators, A loaded as b64.
// Grid covers rows exactly -> EXEC all-ones for every WMMA.
// ---------------------------------------------------------------------------
__global__ void shortcut_mish_wmma_kernel(const float* __restrict__ x,
                                          const float* __restrict__ Kmat,
                                          const float* __restrict__ bias,
                                          const float* __restrict__ yrec,
                                          float* __restrict__ out) {
    __shared__ float sK[CH * CH];      // [k*64 + o], coalesced copy
    const int tid = threadIdx.x;
#pragma unroll
    for (int j = 0; j < (CH * CH) / 256; ++j)
        sK[tid + j * 256] = Kmat[tid + j * 256];
    __syncthreads();

    const int wave = tid >> 5;
    const int lane = tid & 31;
    const int r0   = (blockIdx.x * 8 + wave) * 16;   // first of 16 rows (< 262144)
    const int l15   = lane & 15;
    const int khalf = (lane >> 4) * 2;
    const float* __restrict__ xrow = x + (r0 + l15) * CH;   // per-lane A base

    v8f acc[4] = {{}, {}, {}, {}};
#pragma unroll
    for (int k0 = 0; k0 < CH; k0 += 4) {
        const int ka = k0 + khalf;                          // even -> 8B aligned
        const v2f a = *(const v2f*)&xrow[ka];               // global_load_b64
#pragma unroll
        for (int ct = 0; ct < 4; ++ct) {
            const int col = ct * 16 + l15;
            v2f bf;
            bf.x = sK[ka * CH + col];
            bf.y = sK[(ka + 1) * CH + col];
            acc[ct] = __builtin_amdgcn_wmma_f32_16x16x4_f32(false, a, false, bf,
                                                            (short)0, acc[ct], false, false);
        }
    }

    const int mbase = (lane >> 4) * 8;
#pragma unroll
    for (int ct = 0; ct < 4; ++ct) {
        const int col = ct * 16 + l15;
        const float bv = bias[col];
#pragma unroll
        for (int r = 0; r < 8; ++r) {
            const int row = r0 + mbase + r;
            const int b   = row >> 13;            // N = 8192
            const int nn  = row & (NLEN - 1);
            float z = acc[ct][r] + bv + yrec[(b * CH + col) * NLEN + nn];
            float sp = (z > 20.f) ? z : log1pf(expf(z));   // softplus
            out[row * CH + col] = z * tanhf(sp);           // mish
        }
    }
}

// ---------------------------------------------------------------------------
// Host launcher
// ---------------------------------------------------------------------------
extern "C" void kernel_launch(void* const* d_in, const int* in_sizes, int n_in,
                              void* d_out, int out_size, void* d_ws, size_t ws_size,
                              hipStream_t stream) {
    (void)in_sizes; (void)n_in; (void)out_size; (void)ws_size;
    const float* x     = (const float*)d_in[0];   // (B,N,C)
    const float* w1    = (const float*)d_in[1];   // (C,C,M4)
    const float* w2    = (const float*)d_in[2];   // (C,C,M4)
    const float* dkern = (const float*)d_in[3];   // (C,C)
    const float* dbias = (const float*)d_in[4];   // (C,)
    float* out = (float*)d_out;

    // Workspace slabs (floats). Synthesis reuses analysis-approximation slots.
    float* ws = (float*)d_ws;
    size_t off = 0;
    float* xc  = ws + off; off += (size_t)MROWS * NLEN;   // also holds final r0
    float* a1  = ws + off; off += (size_t)MROWS * M1;     // also holds r1
    float* a2  = ws + off; off += (size_t)MROWS * M2;     // also holds r2
    float* a3  = ws + off; off += (size_t)MROWS * M3;     // also holds r3
    float* a4  = ws + off; off += (size_t)MROWS * M4;
    float* dd1 = ws + off; off += (size_t)MROWS * M1;
    float* dd2 = ws + off; off += (size_t)MROWS * M2;
    float* dd3 = ws + off; off += (size_t)MROWS * M3;
    float* dd4 = ws + off; off += (size_t)MROWS * M4;
    float* a4m = ws + off; off += (size_t)MROWS * M4;
    float* d4m = ws + off; off += (size_t)MROWS * M4;     // total ~193 MiB

    const int TB = 256;
    auto nblk = [](int total, int tb) { return (unsigned)((total + tb - 1) / tb); };

    // 1) layout transform (B,N,C) -> (B*C, N)
    transpose_in_kernel<<<dim3(NLEN / 32, CH / 32, BATCH), dim3(32, 8, 1), 0, stream>>>(x, xc);

    // 2) analysis: 4 DWT levels
    dwt_kernel<<<nblk(MROWS * M1, TB), TB, 0, stream>>>(xc, a1, dd1, NLEN, M1);
    dwt_kernel<<<nblk(MROWS * M2, TB), TB, 0, stream>>>(a1, a2, dd2, M1, M2);
    dwt_kernel<<<nblk(MROWS * M3, TB), TB, 0, stream>>>(a2, a3, dd3, M2, M3);
    dwt_kernel<<<nblk(MROWS * M4, TB), TB, 0, stream>>>(a3, a4, dd4, M3, M4);

    // 3) WMMA channel mixing on coarsest approx / detail (m = M4 compile-time)
    mix_wmma_kernel<<<M4, 256, 0, stream>>>(a4, w1, a4m);
    mix_wmma_kernel<<<M4, 256, 0, stream>>>(dd4, w2, d4m);

    // 4) synthesis: 4 IDWT levels (reusing approximation slots)
    idwt_kernel<<<nblk(MROWS * M3, TB), TB, 0, stream>>>(a4m, d4m, M4, a3, M3);
    idwt_kernel<<<nblk(MROWS * M2, TB), TB, 0, stream>>>(a3, dd3, M3, a2, M2);
    idwt_kernel<<<nblk(MROWS * M1, TB), TB, 0, stream>>>(a2, dd2, M2, a1, M1);
    idwt_kernel<<<nblk(MROWS * NLEN, TB), TB, 0, stream>>>(a1, dd1, M1, xc, NLEN);

    // 5) fused WMMA dense shortcut + reconstruction add + mish -> d_out
    //    262144 rows / (8 waves * 16 rows) = 2048 blocks exactly (EXEC all ones)
    shortcut_mish_wmma_kernel<<<(BATCH * NLEN) / (8 * 16), 256, 0, stream>>>(
        x, dkern, dbias, xc, out);
}